// GTLayer_90263032693118
// MI455X (gfx1250) — compile-verified
//
#include <hip/hip_runtime.h>
#include <hip/hip_bf16.h>

typedef float v2f __attribute__((ext_vector_type(2)));
typedef float v8f __attribute__((ext_vector_type(8)));
typedef unsigned u32x4 __attribute__((ext_vector_type(4)));
typedef int i32x4 __attribute__((ext_vector_type(4)));
typedef int i32x8 __attribute__((ext_vector_type(8)));

#define NNODE 65536
#define NEDGE 524288

// ---------------- TDM availability probing (compile-safe fallbacks) ----------------
#if defined(__has_builtin)
#if __has_builtin(__builtin_amdgcn_tensor_load_to_lds) && __has_builtin(__builtin_amdgcn_s_wait_tensorcnt)
#define HAVE_TDM 1
#endif
#endif

#ifdef HAVE_TDM
__device__ __forceinline__ unsigned lds_off(void* p) {
  return (unsigned)(unsigned long long)(__attribute__((address_space(3))) void*)p;
}
// 1-D contiguous DMA of n dwords: global gsrc -> LDS ldst (issued by one wave).
__device__ __forceinline__ void tdm_copy_1d(const void* gsrc, void* ldst, unsigned n) {
  unsigned long long g = (unsigned long long)gsrc;
  u32x4 g0;
  g0[0] = 1u;                                   // count=1 (valid), no gather
  g0[1] = lds_off(ldst);                        // lds_addr (bytes)
  g0[2] = (unsigned)g;                          // global_addr[31:0]
  g0[3] = (unsigned)((g >> 32) & 0x1ffffffu) | 0x80000000u; // addr[56:32] | type=2
  i32x8 g1;
  g1[0] = 0x00020000;                           // workgroup_mask=0, data_size=2 (4B)
  g1[1] = (int)((n & 0xffffu) << 16);           // tensor_dim0[15:0]
  g1[2] = (int)(((n >> 16) & 0xffffu) | (1u << 16)); // tensor_dim0[31:16], tensor_dim1=1
  g1[3] = (int)(n << 16);                       // tile_dim0 = n (fits 16 bits)
  g1[4] = 0;                                    // tile_dim1=0 (unused), tile_dim2=0
  g1[5] = (int)n;                               // tensor_dim0_stride[31:0]
  g1[6] = 0;
  g1[7] = 0;
  i32x4 z4 = {};
#if __has_include(<hip/amd_detail/amd_gfx1250_TDM.h>)
  i32x8 z8 = {};
  __builtin_amdgcn_tensor_load_to_lds(g0, g1, z4, z4, z8, 0);   // therock 6-arg form
#else
  __builtin_amdgcn_tensor_load_to_lds(g0, g1, z4, z4, 0);       // ROCm 7.2 5-arg form
#endif
}
__device__ __forceinline__ void tdm_wait0() { __builtin_amdgcn_s_wait_tensorcnt(0); }
#endif

// ---- order-preserving float <-> uint encoding for atomicMax-based segment max ----
__device__ __forceinline__ unsigned enc_f32(float x) {
  unsigned u = __float_as_uint(x);
  return (u & 0x80000000u) ? ~u : (u | 0x80000000u);
}
__device__ __forceinline__ float dec_f32(unsigned u) {
  unsigned v = (u & 0x80000000u) ? (u & 0x7fffffffu) : ~u;
  return __uint_as_float(v);
}

// =====================================================================
// Generic fp32 WMMA GEMM:  out[M x NOUT] = epi( LN?(A)[M x K] @ W + bias )
// One wave computes a 16 x NOUT strip using V_WMMA_F32_16X16X4_F32.
// W is staged via the Tensor Data Mover (wave 0) when available.
// EPI: 0 = none, 1 = relu, 2 = gated residual (g*o+(1-g)*aux), 3 = +aux
// =====================================================================
template <int K, int NOUT, bool DO_LN, int EPI, int WAVES>
__global__ __launch_bounds__(WAVES * 32)
void gemm_f32(const float* __restrict__ A,
              const float* __restrict__ ln_g, const float* __restrict__ ln_b,
              const float* __restrict__ W, const float* __restrict__ bias,
              const float* __restrict__ aux, const float* __restrict__ skip,
              float* __restrict__ out)
{
  constexpr int KP = K + 4;                 // padded LDS row stride (bank-conflict free)
  __shared__ float sW[K * NOUT];
  __shared__ float sA[WAVES][16 * KP];

  const int t    = threadIdx.x;
  const int wave = t >> 5;
  const int lane = t & 31;
  const int l16  = lane & 15;
  const int half = lane >> 4;

  // ---- stage W into LDS: TDM DMA (one wave) or coalesced b128 fallback ----
#ifdef HAVE_TDM
  if (wave == 0) tdm_copy_1d(W, sW, K * NOUT);
#else
  for (int i = t * 4; i < K * NOUT; i += WAVES * 32 * 4)
    *(float4*)&sW[i] = *(const float4*)&W[i];
#endif

  // ---- stage this wave's 16xK A tile into padded LDS (source contiguous) ----
  const long long row0 = ((long long)blockIdx.x * WAVES + wave) * 16;
  const float* Ap = A + row0 * K;
  float* sAp = &sA[wave][0];
  for (int i = lane; i < 16 * K; i += 32) {
    int r = i / K, c = i % K;
    sAp[r * KP + c] = Ap[i];
  }
#ifdef HAVE_TDM
  if (wave == 0) tdm_wait0();               // TENSORcnt is per-wave
#endif
  __syncthreads();

  // ---- optional fused LayerNorm over the K columns of each row ----
  if (DO_LN) {
    if (lane < 16) {
      float* rp = sAp + l16 * KP;
      float m = 0.f;
      #pragma unroll
      for (int c = 0; c < K; ++c) m += rp[c];
      m *= (1.0f / K);
      float vv = 0.f;
      #pragma unroll
      for (int c = 0; c < K; ++c) { float d = rp[c] - m; vv += d * d; }
      vv *= (1.0f / K);
      float rs = rsqrtf(vv + 1e-5f);
      #pragma unroll
      for (int c = 0; c < K; ++c) rp[c] = (rp[c] - m) * rs * ln_g[c] + ln_b[c];
    }
    __syncthreads();
  }

  // ---- WMMA main loop: D = A x W, f32 16x16x4 ----
  constexpr int NT = NOUT / 16;
  v8f zero = {};
  v8f acc[NT];
  #pragma unroll
  for (int nt = 0; nt < NT; ++nt) acc[nt] = zero;

  const int kd0 = half ? 2 : 0;             // A/B lane-half K offsets {0,1}/{2,3}
  #pragma unroll
  for (int kk = 0; kk < K; kk += 4) {
    v2f a = *(const v2f*)&sAp[l16 * KP + kk + kd0];   // 8B-aligned -> ds_load_b64
    #pragma unroll
    for (int nt = 0; nt < NT; ++nt) {
      v2f b;
      b.x = sW[(kk + kd0 + 0) * NOUT + nt * 16 + l16];
      b.y = sW[(kk + kd0 + 1) * NOUT + nt * 16 + l16];
      acc[nt] = __builtin_amdgcn_wmma_f32_16x16x4_f32(
          false, a, false, b, (short)0, acc[nt], false, false);
    }
  }

  // ---- epilogue + store (C layout: VGPR r -> rows r / r+8, col = n0+l16) ----
  float gate = 0.f;
  if (EPI == 2) { float s = skip[0]; gate = 1.f / (1.f + __expf(-s)); }
  #pragma unroll
  for (int nt = 0; nt < NT; ++nt) {
    int col = nt * 16 + l16;
    float bc = bias[col];
    #pragma unroll
    for (int r = 0; r < 8; ++r) {
      long long row = row0 + r + 8 * half;
      float val = acc[nt][r] + bc;
      if (EPI == 1) val = fmaxf(val, 0.f);
      else if (EPI == 2) { float xv = aux[row * NOUT + col]; val = gate * val + (1.f - gate) * xv; }
      else if (EPI == 3) { val += aux[row * NOUT + col]; }
      out[row * NOUT + col] = val;
    }
  }
}

// =====================================================================
// Edge pass 1: score = <q_i, (k_j @ edge_w[h]) * e> * scale + bias[h,0]
//              store raw score; atomicMax per (dst,h)
// =====================================================================
__global__ __launch_bounds__(256)
void edge_score_kernel(const int* __restrict__ ei,
                       const float* __restrict__ q, const float* __restrict__ k,
                       const float* __restrict__ eproj,
                       const float* __restrict__ edge_w,
                       const float* __restrict__ attn_bi,
                       float* __restrict__ wbuf, unsigned* __restrict__ nmax)
{
  __shared__ float sEW[4 * 256];
  for (int i = threadIdx.x; i < 1024; i += 256) sEW[i] = edge_w[i];
  __syncthreads();

  int t = blockIdx.x * 256 + threadIdx.x;
  int e = t >> 2;
  int h = t & 3;
  int src = ei[e];
  int dst = ei[NEDGE + e];

  const float* qp = q + dst * 64 + h * 16;
  const float* kp = k + src * 64 + h * 16;
  const float* ep = eproj + e * 64 + h * 16;

  float kj[16], qe[16];
  #pragma unroll
  for (int d = 0; d < 16; ++d) { kj[d] = kp[d]; qe[d] = qp[d] * ep[d]; }

  const float* ew = sEW + h * 256;
  float s = 0.f;
  #pragma unroll
  for (int c = 0; c < 16; ++c) {
    float kr = 0.f;
    #pragma unroll
    for (int d = 0; d < 16; ++d) kr += kj[d] * ew[d * 16 + c];
    s += qe[c] * kr;
  }
  s = s * 0.25f + attn_bi[h * 2];     // scale = 1/sqrt(16), 1-hop bias
  wbuf[e * 4 + h] = s;
  atomicMax(&nmax[dst * 4 + h], enc_f32(s));
}

// =====================================================================
// Edge pass 2: w = exp(score - max[dst,h]); atomicAdd sum[dst,h]
// =====================================================================
__global__ __launch_bounds__(256)
void edge_exp_kernel(const int* __restrict__ ei,
                     float* __restrict__ wbuf,
                     const unsigned* __restrict__ nmax,
                     float* __restrict__ nsum)
{
  int t = blockIdx.x * 256 + threadIdx.x;
  int e = t >> 2;
  int h = t & 3;
  int dst = ei[NEDGE + e];
  float m  = dec_f32(nmax[dst * 4 + h]);
  float wv = __expf(wbuf[e * 4 + h] - m);
  wbuf[e * 4 + h] = wv;
  unsafeAtomicAdd(&nsum[dst * 4 + h], wv);
}

// =====================================================================
// Edge pass 3: alpha = w / (sum+eps);  attn[dst] += alpha * (v_j @ msg_w[h])
// =====================================================================
__global__ __launch_bounds__(256)
void edge_scatter_kernel(const int* __restrict__ ei,
                         const float* __restrict__ v,
                         const float* __restrict__ msg_w,
                         const float* __restrict__ wbuf,
                         const float* __restrict__ nsum,
                         float* __restrict__ attn)
{
  __shared__ float sMW[4 * 256];
  for (int i = threadIdx.x; i < 1024; i += 256) sMW[i] = msg_w[i];
  __syncthreads();

  int t = blockIdx.x * 256 + threadIdx.x;
  int e = t >> 2;
  int h = t & 3;
  int src = ei[e];
  int dst = ei[NEDGE + e];

  float alpha = wbuf[e * 4 + h] / (nsum[dst * 4 + h] + 1e-16f);

  const float* vp = v + src * 64 + h * 16;
  float vj[16];
  #pragma unroll
  for (int d = 0; d < 16; ++d) vj[d] = vp[d];

  const float* mw = sMW + h * 256;
  float* op = attn + dst * 64 + h * 16;
  #pragma unroll
  for (int c = 0; c < 16; ++c) {
    float vr = 0.f;
    #pragma unroll
    for (int d = 0; d < 16; ++d) vr += vj[d] * mw[d * 16 + c];
    unsafeAtomicAdd(&op[c], alpha * vr);
  }
}

// =====================================================================
extern "C" void kernel_launch(void* const* d_in, const int* in_sizes, int n_in,
                              void* d_out, int out_size, void* d_ws, size_t ws_size,
                              hipStream_t stream) {
  const float* x     = (const float*)d_in[0];
  const int*   ei    = (const int*)  d_in[1];
  const float* eattr = (const float*)d_in[2];
  const float* Wq = (const float*)d_in[3];  const float* bq = (const float*)d_in[4];
  const float* Wk = (const float*)d_in[5];  const float* bk = (const float*)d_in[6];
  const float* Wv = (const float*)d_in[7];  const float* bv = (const float*)d_in[8];
  const float* We = (const float*)d_in[9];  const float* be = (const float*)d_in[10];
  const float* Wo = (const float*)d_in[11]; const float* bo = (const float*)d_in[12];
  const float* edge_w = (const float*)d_in[13];
  const float* msg_w  = (const float*)d_in[14];
  const float* skip   = (const float*)d_in[15];
  const float* ln1_g = (const float*)d_in[16]; const float* ln1_b = (const float*)d_in[17];
  const float* lne_g = (const float*)d_in[18]; const float* lne_b = (const float*)d_in[19];
  const float* ln2_g = (const float*)d_in[20]; const float* ln2_b = (const float*)d_in[21];
  const float* W1 = (const float*)d_in[22]; const float* b1 = (const float*)d_in[23];
  const float* W2 = (const float*)d_in[24]; const float* b2 = (const float*)d_in[25];
  const float* attn_bi = (const float*)d_in[26];

  // ---- workspace layout ----
  float* f = (float*)d_ws;
  float* qb    = f; f += (size_t)NNODE * 64;
  float* kb    = f; f += (size_t)NNODE * 64;
  float* vb    = f; f += (size_t)NNODE * 64;
  float* eproj = f; f += (size_t)NEDGE * 64;
  float* wbuf  = f; f += (size_t)NEDGE * 4;
  float*    nsum = f;            f += (size_t)NNODE * 4;
  unsigned* nmax = (unsigned*)f; f += (size_t)NNODE * 4;
  float* attn  = f; f += (size_t)NNODE * 64;
  float* hres  = f; f += (size_t)NNODE * 64;
  float* ffnt  = f; f += (size_t)NNODE * 128;

  // zero nsum + nmax (0u == below every encoded float) + attn accumulator
  hipMemsetAsync(nsum, 0, (size_t)NNODE * (4 + 4 + 64) * sizeof(float), stream);

  // ---- fused LN + QKV projections (WMMA f32) ----
  gemm_f32<64, 64, true, 0, 4><<<NNODE / 64, 128, 0, stream>>>(
      x, ln1_g, ln1_b, Wq, bq, nullptr, nullptr, qb);
  gemm_f32<64, 64, true, 0, 4><<<NNODE / 64, 128, 0, stream>>>(
      x, ln1_g, ln1_b, Wk, bk, nullptr, nullptr, kb);
  gemm_f32<64, 64, true, 0, 4><<<NNODE / 64, 128, 0, stream>>>(
      x, ln1_g, ln1_b, Wv, bv, nullptr, nullptr, vb);

  // ---- fused LN + edge projection (biggest GEMM: E x 64 x 64) ----
  gemm_f32<64, 64, true, 0, 4><<<NEDGE / 64, 128, 0, stream>>>(
      eattr, lne_g, lne_b, We, be, nullptr, nullptr, eproj);

  // ---- segment softmax over destination nodes (3 atomic passes) ----
  int egrid = (NEDGE * 4) / 256;
  edge_score_kernel<<<egrid, 256, 0, stream>>>(ei, qb, kb, eproj, edge_w, attn_bi, wbuf, nmax);
  edge_exp_kernel<<<egrid, 256, 0, stream>>>(ei, wbuf, nmax, nsum);
  edge_scatter_kernel<<<egrid, 256, 0, stream>>>(ei, vb, msg_w, wbuf, nsum, attn);

  // ---- output projection + gated residual: h = g*(attn@Wo+bo) + (1-g)*x ----
  gemm_f32<64, 64, false, 2, 4><<<NNODE / 64, 128, 0, stream>>>(
      attn, nullptr, nullptr, Wo, bo, x, skip, hres);

  // ---- FFN: relu(LN(h)@W1+b1) ----
  gemm_f32<64, 128, true, 1, 4><<<NNODE / 64, 128, 0, stream>>>(
      hres, ln2_g, ln2_b, W1, b1, nullptr, nullptr, ffnt);

  // ---- FFN out + residual -> d_out ----
  gemm_f32<128, 64, false, 3, 2><<<NNODE / 32, 64, 0, stream>>>(
      ffnt, nullptr, nullptr, W2, b2, hres, nullptr, (float*)d_out);
}